// DefuzzyLayer2_8160437862725
// MI455X (gfx1250) — compile-verified
//
#include <hip/hip_runtime.h>

// out[b,o] = sum_d x^2[b,d]*W2[d,o] + sum_d x[b,d]*W1[d,o] + colsum(bias)[o]
// Two fused fp32 GEMMs on the CDNA5 matrix pipe via V_WMMA_F32_16X16X4_F32.

#define D_IN    512
#define D_OUT   512
#define NBATCH  8192

#define BLOCK_M 256      // rows of x per workgroup
#define BLOCK_N 64       // cols of out per workgroup
#define BK      16       // K tile staged in LDS
#define BKP     20       // padded K stride (floats): bank = (20*m + k) & 63 -> conflict-free b64 frags
#define KTILES  (D_IN / BK)   // 32

typedef float v2f __attribute__((ext_vector_type(2)));
typedef float v8f __attribute__((ext_vector_type(8)));

// ---------------------------------------------------------------------------
// Kernel 1: colsum[o] = sum_d bias[d][o]  (coalesced across threads)
// ---------------------------------------------------------------------------
__global__ void __launch_bounds__(256) bias_colsum_kernel(
    const float* __restrict__ bias, float* __restrict__ colsum) {
  const int o = blockIdx.x * blockDim.x + threadIdx.x;   // 0..511
  float s = 0.f;
#pragma unroll 8
  for (int d = 0; d < D_IN; ++d) s += bias[(size_t)d * D_OUT + o];
  colsum[o] = s;
}

// ---------------------------------------------------------------------------
// Kernel 2: fused double-GEMM with WMMA f32 16x16x4
//   8 waves/block; wave w owns rows [w*32, w*32+32) x all 64 cols
//   LDS layouts are K-contiguous so every A/B fragment is one ds_load_b64:
//     lane L holds (M or N) = L%16, K = (L/16)*2 + vgpr  -> per-lane consecutive K
// ---------------------------------------------------------------------------
__global__ void __launch_bounds__(256) defuzzy_wmma_kernel(
    const float* __restrict__ x,  const float* __restrict__ w1,
    const float* __restrict__ w2, const float* __restrict__ colsum,
    float* __restrict__ out) {
  __shared__ float Xs[2][BLOCK_M][BKP];       // 40 KB  [m][k]
  __shared__ float Ws[2][2][BLOCK_N][BKP];    // 20 KB  [mat][n][k] (transposed vs global)

  const int tid     = threadIdx.x;
  const int lane    = tid & 31;
  const int wave    = tid >> 5;
  const int lane_lo = lane & 15;
  const int kofs    = (lane >> 4) * 2;        // K offset of this half-wave in a 4-wide frag

  const int m_base = blockIdx.x * BLOCK_M;
  const int n_base = blockIdx.y * BLOCK_N;

  v8f acc[2][4];
#pragma unroll
  for (int s = 0; s < 2; ++s)
#pragma unroll
    for (int nt = 0; nt < 4; ++nt) acc[s][nt] = {};

  float4 xr[4];
  float  w1r[4], w2r[4];

  // ---- prologue: stage tile 0 into buffer 0
  {
#pragma unroll
    for (int i = 0; i < 4; ++i) {
      const int linear = tid + i * 256;
      const int row = linear >> 2, c4 = linear & 3;          // x: float4 per thread
      xr[i] = *(const float4*)&x[(size_t)(m_base + row) * D_IN + c4 * 4];
      const int n = linear & 63, kr = linear >> 6;           // W: b32 per thread, coalesced in n
      w1r[i] = w1[(size_t)kr * D_OUT + n_base + n];
      w2r[i] = w2[(size_t)kr * D_OUT + n_base + n];
    }
#pragma unroll
    for (int i = 0; i < 4; ++i) {
      const int linear = tid + i * 256;
      const int row = linear >> 2, c4 = linear & 3;
      *(float4*)&Xs[0][row][c4 * 4] = xr[i];                 // 16B-aligned (80*row + 16*c4)
      const int n = linear & 63, kr = linear >> 6;
      Ws[0][0][n][kr] = w1r[i];                              // transpose [k][n] -> [n][k]
      Ws[0][1][n][kr] = w2r[i];
    }
  }
  __syncthreads();

  for (int t = 0; t < KTILES; ++t) {
    const int  cur      = t & 1;
    const bool has_next = (t + 1) < KTILES;

    // ---- issue global loads for next tile (overlap with compute below)
    if (has_next) {
      const int k0 = (t + 1) * BK;
#pragma unroll
      for (int i = 0; i < 4; ++i) {
        const int linear = tid + i * 256;
        const int row = linear >> 2, c4 = linear & 3;
        xr[i] = *(const float4*)&x[(size_t)(m_base + row) * D_IN + k0 + c4 * 4];
        const int n = linear & 63, kr = linear >> 6;
        w1r[i] = w1[(size_t)(k0 + kr) * D_OUT + n_base + n];
        w2r[i] = w2[(size_t)(k0 + kr) * D_OUT + n_base + n];
      }
    }

    // ---- compute on current buffer: 4 K-steps x 16 WMMAs
#pragma unroll
    for (int kk = 0; kk < BK; kk += 4) {
      v2f a[2], a2[2];
#pragma unroll
      for (int s = 0; s < 2; ++s) {
        const int m = wave * 32 + s * 16 + lane_lo;
        a[s]  = *(const v2f*)&Xs[cur][m][kk + kofs];         // one ds_load_b64
        a2[s] = a[s] * a[s];                                 // x^2 fragment (VALU)
      }
#pragma unroll
      for (int nt = 0; nt < 4; ++nt) {
        const int n = nt * 16 + lane_lo;
        const v2f b1 = *(const v2f*)&Ws[cur][0][n][kk + kofs];
        const v2f b2 = *(const v2f*)&Ws[cur][1][n][kk + kofs];
#pragma unroll
        for (int s = 0; s < 2; ++s) {
          acc[s][nt] = __builtin_amdgcn_wmma_f32_16x16x4_f32(
              false, a[s],  false, b1, (short)0, acc[s][nt], false, false);
          acc[s][nt] = __builtin_amdgcn_wmma_f32_16x16x4_f32(
              false, a2[s], false, b2, (short)0, acc[s][nt], false, false);
        }
      }
    }
    __syncthreads();   // all waves done reading `cur` before it is refilled at t+1

    if (has_next) {
      const int nb = (t + 1) & 1;
#pragma unroll
      for (int i = 0; i < 4; ++i) {
        const int linear = tid + i * 256;
        const int row = linear >> 2, c4 = linear & 3;
        *(float4*)&Xs[nb][row][c4 * 4] = xr[i];
        const int n = linear & 63, kr = linear >> 6;
        Ws[nb][0][n][kr] = w1r[i];
        Ws[nb][1][n][kr] = w2r[i];
      }
      __syncthreads();
    }
  }

  // ---- epilogue: add bias column-sum, store (half-wave rows are 64B-contiguous)
#pragma unroll
  for (int nt = 0; nt < 4; ++nt) {
    const int   col = n_base + nt * 16 + lane_lo;
    const float bs  = colsum[col];
#pragma unroll
    for (int s = 0; s < 2; ++s) {
      const int row0 = m_base + wave * 32 + s * 16 + (lane >> 4) * 8;
#pragma unroll
      for (int j = 0; j < 8; ++j)   // C/D layout: VGPR j = M rows j (lanes 0-15) / j+8 (16-31)
        out[(size_t)(row0 + j) * D_OUT + col] = acc[s][nt][j] + bs;
    }
  }
}

// ---------------------------------------------------------------------------
extern "C" void kernel_launch(void* const* d_in, const int* in_sizes, int n_in,
                              void* d_out, int out_size, void* d_ws, size_t ws_size,
                              hipStream_t stream) {
  const float* x    = (const float*)d_in[0];  // [8192, 512]
  const float* w1   = (const float*)d_in[1];  // rules_outcome   [512, 512]
  const float* bias = (const float*)d_in[2];  // bias            [512, 512]
  const float* w2   = (const float*)d_in[3];  // rules_outcome_2 [512, 512]
  float* out    = (float*)d_out;              // [8192, 512]
  float* colsum = (float*)d_ws;               // 512 floats of scratch

  bias_colsum_kernel<<<dim3(D_OUT / 256), dim3(256), 0, stream>>>(bias, colsum);

  dim3 grid(NBATCH / BLOCK_M, D_OUT / BLOCK_N);   // (32, 8)
  defuzzy_wmma_kernel<<<grid, dim3(256), 0, stream>>>(x, w1, w2, colsum, out);
}